// UpsamplingAttention_16655883174384
// MI455X (gfx1250) — compile-verified
//
#include <hip/hip_runtime.h>
#include <math.h>

#define N_   16
#define K_   256
#define T_   1000
#define M_   384
#define TPAD 1024          // T padded to multiple of 32 for 2x2 WMMA register blocking
#define BN_EPS 1e-3f

typedef __attribute__((ext_vector_type(2))) float v2f;
typedef __attribute__((ext_vector_type(8))) float v8f;

__device__ __forceinline__ float silu_f(float x) { return x / (1.0f + __expf(-x)); }

// ---------------------------------------------------------------------------
// Kernel A: durations = relu(exp(d)-1); S = exclusive cumsum; E = inclusive.
// One block per n, 256 threads (one per k). Hillis-Steele scan in LDS.
// ---------------------------------------------------------------------------
__global__ void dur_scan_kernel(const float* __restrict__ dur,
                                float* __restrict__ S, float* __restrict__ E) {
    int n = blockIdx.x, k = threadIdx.x;
    __shared__ float sd[K_];
    float d = __expf(dur[n * K_ + k]) - 1.0f;
    d = d > 0.0f ? d : 0.0f;
    sd[k] = d;
    for (int off = 1; off < K_; off <<= 1) {
        __syncthreads();
        float v = (k >= off) ? sd[k - off] : 0.0f;
        __syncthreads();
        sd[k] += v;
    }
    float incl = sd[k];
    S[n * K_ + k] = incl - d;
    E[n * K_ + k] = incl;
}

// ---------------------------------------------------------------------------
// Kernel B: conv1d(M->8, ksize 3, SAME) + bias + BN(inference) + SiLU.
// One thread per (n,k,o). convV layout [N][K][8].
// ---------------------------------------------------------------------------
__global__ void conv_bn_silu_kernel(const float* __restrict__ V,
                                    const float* __restrict__ cw,
                                    const float* __restrict__ cb,
                                    const float* __restrict__ g,
                                    const float* __restrict__ be,
                                    const float* __restrict__ mu,
                                    const float* __restrict__ var,
                                    float* __restrict__ convV) {
    int idx = blockIdx.x * blockDim.x + threadIdx.x;
    if (idx >= N_ * K_ * 8) return;
    int o = idx & 7;
    int k = (idx >> 3) & (K_ - 1);
    int n = idx >> 11;                       // K_*8 == 2048
    float acc = cb[o];
    for (int dk = -1; dk <= 1; ++dk) {
        int kk = k + dk;
        if (kk < 0 || kk >= K_) continue;
        const float* vp = V + ((size_t)n * K_ + kk) * M_;
        const float* wp = cw + (size_t)(dk + 1) * M_ * 8 + o;
#pragma unroll 4
        for (int c = 0; c < M_; ++c) acc = fmaf(vp[c], wp[c * 8], acc);
    }
    float sc = g[o] * rsqrtf(var[o] + BN_EPS);
    float y  = (acc - mu[o]) * sc + be[o];
    convV[idx] = silu_f(y);
}

// ---------------------------------------------------------------------------
// Kernel C: per (n,t): build 10-feature vector per k, score MLP (10->16->16->1),
// softmax over K, aux MLP (10->2->2), wc[p] = sum_k W*C. One block per (n,t'),
// t' in [0,TPAD): pad rows write zero W/wc so the GEMM is unguarded on A.
// ---------------------------------------------------------------------------
__global__ void score_softmax_kernel(const float* __restrict__ S,
                                     const float* __restrict__ E,
                                     const float* __restrict__ convV,
                                     const float* __restrict__ w1, const float* __restrict__ b1,
                                     const float* __restrict__ w2, const float* __restrict__ b2,
                                     const float* __restrict__ w3, const float* __restrict__ b3,
                                     const float* __restrict__ aw1, const float* __restrict__ ab1,
                                     const float* __restrict__ aw2, const float* __restrict__ ab2,
                                     float* __restrict__ Wbuf, float* __restrict__ wcbuf) {
    int t = blockIdx.x, n = blockIdx.y, k = threadIdx.x;
    float* Wrow = Wbuf + ((size_t)n * TPAD + t) * K_;
    if (t >= T_) {                            // zero-fill pad rows every call
        Wrow[k] = 0.0f;
        if (k < 2) wcbuf[((size_t)n * TPAD + t) * 2 + k] = 0.0f;
        return;
    }
    __shared__ float sw1[160], sb1[16], sw2[256], sb2[16], sw3[16], saux[28], red[K_];
    __shared__ float sb3s;
    sw2[k] = w2[k];
    if (k < 160) sw1[k] = w1[k];
    if (k < 16) { sb1[k] = b1[k]; sb2[k] = b2[k]; sw3[k] = w3[k]; }
    if (k < 20) saux[k] = aw1[k];            // aw1: 10x2
    if (k < 4)  saux[22 + k] = aw2[k];       // aw2: 2x2
    if (k < 2)  { saux[20 + k] = ab1[k]; saux[26 + k] = ab2[k]; }
    if (k == 0) sb3s = b3[0];
    __syncthreads();

    float tv = (float)(t + 1);
    float x[10];
    x[0] = tv - S[n * K_ + k];
    x[1] = E[n * K_ + k] - tv;
    const float* cv = convV + ((size_t)n * K_ + k) * 8;
#pragma unroll
    for (int i = 0; i < 8; ++i) x[2 + i] = cv[i];

    float h1[16];
#pragma unroll
    for (int j = 0; j < 16; ++j) {
        float a = sb1[j];
#pragma unroll
        for (int i = 0; i < 10; ++i) a = fmaf(x[i], sw1[i * 16 + j], a);
        h1[j] = silu_f(a);
    }
    float h2[16];
#pragma unroll
    for (int j = 0; j < 16; ++j) {
        float a = sb2[j];
#pragma unroll
        for (int i = 0; i < 16; ++i) a = fmaf(h1[i], sw2[i * 16 + j], a);
        h2[j] = silu_f(a);
    }
    float score = sb3s;
#pragma unroll
    for (int i = 0; i < 16; ++i) score = fmaf(h2[i], sw3[i], score);

    // aux MLP 10 -> 2 -> 2
    float a1[2], a2[2];
#pragma unroll
    for (int p = 0; p < 2; ++p) {
        float a = saux[20 + p];
#pragma unroll
        for (int i = 0; i < 10; ++i) a = fmaf(x[i], saux[i * 2 + p], a);
        a1[p] = silu_f(a);
    }
#pragma unroll
    for (int p = 0; p < 2; ++p) {
        float a = saux[26 + p];
        a = fmaf(a1[0], saux[22 + p], a);
        a = fmaf(a1[1], saux[24 + p], a);
        a2[p] = silu_f(a);
    }

    // softmax over K
    red[k] = score; __syncthreads();
    for (int off = 128; off > 0; off >>= 1) {
        if (k < off) red[k] = fmaxf(red[k], red[k + off]);
        __syncthreads();
    }
    float mx = red[0]; __syncthreads();
    float e = __expf(score - mx);
    red[k] = e; __syncthreads();
    for (int off = 128; off > 0; off >>= 1) {
        if (k < off) red[k] += red[k + off];
        __syncthreads();
    }
    float wk = e / red[0];
    Wrow[k] = wk;
    __syncthreads();

    // wc[p] = sum_k W*C[,p]
    red[k] = wk * a2[0]; __syncthreads();
    for (int off = 128; off > 0; off >>= 1) {
        if (k < off) red[k] += red[k + off];
        __syncthreads();
    }
    float wc0 = red[0]; __syncthreads();
    red[k] = wk * a2[1]; __syncthreads();
    for (int off = 128; off > 0; off >>= 1) {
        if (k < off) red[k] += red[k + off];
        __syncthreads();
    }
    if (k == 0) {
        wcbuf[((size_t)n * TPAD + t) * 2 + 0] = wc0;
        wcbuf[((size_t)n * TPAD + t) * 2 + 1] = red[0];
    }
}

// ---------------------------------------------------------------------------
// Kernel D: per n, O[t,m] = sum_k W[t,k] * V[k,m] + wc[t]·proj_w[:,m]
// f32 WMMA 16x16x4 with 2x2 register blocking: each wave owns a 32(t)x32(m)
// output tile (4 accumulators), so each K-step of 4 issues 4 WMMAs fed by
// only 2 b64 A-loads + 4 b32 B-loads (1.5 loads/WMMA vs 3 for 1x1 tiles).
// A frag (ISA 32-bit 16x4): lane holds row=lane&15, k-pair=(lane>>4)*2.
// C/D frag: vgpr r <-> row r + 8*(lane>>4), col lane&15.
// ---------------------------------------------------------------------------
__global__ void wmma_gemm_kernel(const float* __restrict__ Wbuf,
                                 const float* __restrict__ V,
                                 const float* __restrict__ wcbuf,
                                 const float* __restrict__ proj_w,
                                 float* __restrict__ O) {
    int wave = threadIdx.x >> 5;
    int lane = threadIdx.x & 31;
    int m0 = (blockIdx.x * 4 + wave) * 32;   // 4 waves cover 128 columns
    int t0 = blockIdx.y * 32;
    int n  = blockIdx.z;

    int rc   = lane & 15;                // A row / B col within a 16x16 tile
    int koff = (lane >> 4) * 2;          // which k-pair this half-wave owns

    const float* A0 = Wbuf + ((size_t)n * TPAD + (t0 + rc)) * K_ + koff;
    const float* A1 = A0 + (size_t)16 * K_;
    const float* B0 = V + (size_t)n * K_ * M_ + (size_t)koff * M_ + m0 + rc;

    v8f acc00 = {}, acc01 = {}, acc10 = {}, acc11 = {};
#pragma unroll 2
    for (int kk = 0; kk < K_; kk += 4) {
        v2f a0 = *(const v2f*)(A0 + kk);           // W[t0   +row][kk+koff..+1]
        v2f a1 = *(const v2f*)(A1 + kk);           // W[t0+16+row][kk+koff..+1]
        const float* bp = B0 + (size_t)kk * M_;
        v2f b0, b1;
        b0.x = bp[0];        b0.y = bp[M_];        // V[k][m0+col],    V[k+1][m0+col]
        b1.x = bp[16];       b1.y = bp[M_ + 16];   // V[k][m0+16+col], V[k+1][m0+16+col]
        acc00 = __builtin_amdgcn_wmma_f32_16x16x4_f32(false, a0, false, b0, (short)0, acc00, false, false);
        acc01 = __builtin_amdgcn_wmma_f32_16x16x4_f32(false, a0, false, b1, (short)0, acc01, false, false);
        acc10 = __builtin_amdgcn_wmma_f32_16x16x4_f32(false, a1, false, b0, (short)0, acc10, false, false);
        acc11 = __builtin_amdgcn_wmma_f32_16x16x4_f32(false, a1, false, b1, (short)0, acc11, false, false);
    }

    int col   = lane & 15;
    int rbase = (lane >> 4) * 8;
    float pw00 = proj_w[m0 + col];
    float pw01 = proj_w[M_ + m0 + col];
    float pw10 = proj_w[m0 + 16 + col];
    float pw11 = proj_w[M_ + m0 + 16 + col];
#pragma unroll
    for (int r = 0; r < 8; ++r) {
        int trow = t0 + rbase + r;
        if (trow < T_) {
            v2f wc = *(const v2f*)(wcbuf + ((size_t)n * TPAD + trow) * 2);
            float* op = O + ((size_t)n * T_ + trow) * M_ + m0 + col;
            op[0]  = acc00[r] + wc.x * pw00 + wc.y * pw01;
            op[16] = acc01[r] + wc.x * pw10 + wc.y * pw11;
        }
        int trow1 = trow + 16;
        if (trow1 < T_) {
            v2f wc = *(const v2f*)(wcbuf + ((size_t)n * TPAD + trow1) * 2);
            float* op = O + ((size_t)n * T_ + trow1) * M_ + m0 + col;
            op[0]  = acc10[r] + wc.x * pw00 + wc.y * pw01;
            op[16] = acc11[r] + wc.x * pw10 + wc.y * pw11;
        }
    }
}

// ---------------------------------------------------------------------------
extern "C" void kernel_launch(void* const* d_in, const int* in_sizes, int n_in,
                              void* d_out, int out_size, void* d_ws, size_t ws_size,
                              hipStream_t stream) {
    const float* V      = (const float*)d_in[0];
    const float* dur    = (const float*)d_in[1];
    const float* conv_w = (const float*)d_in[2];
    const float* conv_b = (const float*)d_in[3];
    const float* bn_g   = (const float*)d_in[4];
    const float* bn_b   = (const float*)d_in[5];
    const float* bn_m   = (const float*)d_in[6];
    const float* bn_v   = (const float*)d_in[7];
    const float* w1     = (const float*)d_in[8];
    const float* b1     = (const float*)d_in[9];
    const float* w2     = (const float*)d_in[10];
    const float* b2     = (const float*)d_in[11];
    const float* w3     = (const float*)d_in[12];
    const float* b3     = (const float*)d_in[13];
    const float* aw1    = (const float*)d_in[14];
    const float* ab1    = (const float*)d_in[15];
    const float* aw2    = (const float*)d_in[16];
    const float* ab2    = (const float*)d_in[17];
    const float* proj_w = (const float*)d_in[18];
    float* out = (float*)d_out;

    // workspace layout (floats): S[N*K] | E[N*K] | convV[N*K*8] | W[N*TPAD*K] | wc[N*TPAD*2]
    float* ws    = (float*)d_ws;
    float* S     = ws;
    float* E     = S + N_ * K_;
    float* convV = E + N_ * K_;
    float* Wbuf  = convV + N_ * K_ * 8;
    float* wcbuf = Wbuf + (size_t)N_ * TPAD * K_;

    dur_scan_kernel<<<N_, K_, 0, stream>>>(dur, S, E);

    conv_bn_silu_kernel<<<(N_ * K_ * 8 + 255) / 256, 256, 0, stream>>>(
        V, conv_w, conv_b, bn_g, bn_b, bn_m, bn_v, convV);

    dim3 gC(TPAD, N_);
    score_softmax_kernel<<<gC, K_, 0, stream>>>(
        S, E, convV, w1, b1, w2, b2, w3, b3, aw1, ab1, aw2, ab2, Wbuf, wcbuf);

    dim3 gD(M_ / 128, TPAD / 32, N_);   // (3, 32, 16), 4 waves per block
    wmma_gemm_kernel<<<gD, 128, 0, stream>>>(Wbuf, V, wcbuf, proj_w, out);
}